// LeADMM_54099408060510
// MI455X (gfx1250) — compile-verified
//
#include <hip/hip_runtime.h>
#include <math.h>
#include <stdint.h>

// Le-ADMM (state reset each iter; only last iter's output survives) == one LSI filter:
//   out = crop(fftshift(ifft2( Rdiv * conj(Hf) * fft2(ifftshift(pad(x))*mu1[4]/(1+1e-6)) )))
// Bandwidth-bound on MI455X (23.3 TB/s): LDS-resident mixed-radix Stockham FFTs,
// fused fwd-col -> filter -> inv-col single pass, f32 WMMA (16x16x4) for the
// twiddle-free radix-16 first stage of the 1920-pt rows, TDM tile copies for
// global<->LDS movement.

#define SH 540
#define SW 960
#define FH 1080
#define FW 1920
#define NB 8
#define TC 8
#define MU1_0 1e-6f
#define MU2_0 1e-5f
#define MU3_0 4e-5f
#define PI_F 3.14159265358979323846f

typedef float v2f __attribute__((ext_vector_type(2)));
typedef float v8f __attribute__((ext_vector_type(8)));
typedef unsigned int v4u __attribute__((ext_vector_type(4)));
typedef int v4i __attribute__((ext_vector_type(4)));
typedef int v8i __attribute__((ext_vector_type(8)));

#if defined(__has_builtin)
#if __has_builtin(__builtin_amdgcn_tensor_load_to_lds) && \
    __has_builtin(__builtin_amdgcn_tensor_store_from_lds) && \
    __has_builtin(__builtin_amdgcn_s_wait_tensorcnt)
#define HAVE_TDM 1
#endif
#endif
#if __has_include(<hip/amd_detail/amd_gfx1250_TDM.h>)
#define TDM_SIX_ARGS 1
#endif

// ---------------------------------------------------------------------------
// Tensor Data Mover: 2D tile (tileW x tileH elems, row stride strideElems,
// 4-byte data) between global and LDS.  D# per CDNA5 ISA.
// ---------------------------------------------------------------------------
#ifdef HAVE_TDM
__device__ __forceinline__ void tdm_desc(v4u& g0, v8i& g1, const void* gaddr,
                                         uint32_t ldsOff, uint32_t tileW,
                                         uint32_t tileH, uint32_t strideElems) {
  const uint64_t ga = (uint64_t)(uintptr_t)gaddr;
  g0.x = 1u;                                    // count=1, user descriptor
  g0.y = ldsOff;                                // lds_addr (bytes)
  g0.z = (uint32_t)ga;                          // global_addr[31:0]
  g0.w = (uint32_t)((ga >> 32) & 0x01FFFFFFu) | (2u << 30);  // [56:32] | type=2
  g1[0] = (int)(2u << 16);                      // wg_mask=0, data_size=4B
  g1[1] = (int)((strideElems & 0xFFFFu) << 16);                 // tensor_dim0 lo
  g1[2] = (int)(((strideElems >> 16) & 0xFFFFu) | ((tileH & 0xFFFFu) << 16));
  g1[3] = (int)(((tileH >> 16) & 0xFFFFu) | ((tileW & 0xFFFFu) << 16)); // tile_dim0
  g1[4] = (int)(tileH & 0xFFFFu);               // tile_dim1 (tile_dim2=0)
  g1[5] = (int)strideElems;                     // tensor_dim0_stride lo32
  g1[6] = 0;
  g1[7] = 0;
}

__device__ __forceinline__ void tdm_load_tile(const void* g, uint32_t ldsOff,
                                              uint32_t tw, uint32_t th, uint32_t se) {
  v4u g0; v8i g1;
  tdm_desc(g0, g1, g, ldsOff, tw, th, se);
  v4i z; z.x = 0; z.y = 0; z.z = 0; z.w = 0;
#ifdef TDM_SIX_ARGS
  v8i z8; for (int i = 0; i < 8; ++i) z8[i] = 0;
  __builtin_amdgcn_tensor_load_to_lds(g0, g1, z, z, z8, 0);
#else
  __builtin_amdgcn_tensor_load_to_lds(g0, g1, z, z, 0);
#endif
}

__device__ __forceinline__ void tdm_store_tile(const void* g, uint32_t ldsOff,
                                               uint32_t tw, uint32_t th, uint32_t se) {
  v4u g0; v8i g1;
  tdm_desc(g0, g1, g, ldsOff, tw, th, se);
  v4i z; z.x = 0; z.y = 0; z.z = 0; z.w = 0;
#ifdef TDM_SIX_ARGS
  v8i z8; for (int i = 0; i < 8; ++i) z8[i] = 0;
  __builtin_amdgcn_tensor_store_from_lds(g0, g1, z, z, z8, 0);
#else
  __builtin_amdgcn_tensor_store_from_lds(g0, g1, z, z, 0);
#endif
}
#endif  // HAVE_TDM

__device__ __forceinline__ uint32_t lds_off(const float* base, const float* p) {
  return (uint32_t)((const char*)p - (const char*)base);
}

// ---------------------------------------------------------------------------
// Complex helpers + constant-folded small-DFT butterflies (sign sg = -1 fwd).
// ---------------------------------------------------------------------------
struct cpx { float r, i; };
__device__ __forceinline__ cpx cadd(cpx a, cpx b) { return cpx{a.r + b.r, a.i + b.i}; }
__device__ __forceinline__ cpx csub(cpx a, cpx b) { return cpx{a.r - b.r, a.i - b.i}; }
__device__ __forceinline__ cpx cmulj(cpx z, float sg) { return cpx{-sg * z.i, sg * z.r}; }

__device__ __forceinline__ void bfly(cpx (&a)[3], float sg) {
  const float S3 = 0.86602540378443864676f;
  cpx t = cadd(a[1], a[2]);
  cpx u = csub(a[1], a[2]);
  cpx m = cpx{a[0].r - 0.5f * t.r, a[0].i - 0.5f * t.i};
  const float s = sg * S3;
  a[0] = cadd(a[0], t);
  a[1] = cpx{m.r - s * u.i, m.i + s * u.r};
  a[2] = cpx{m.r + s * u.i, m.i - s * u.r};
}

__device__ __forceinline__ void bfly(cpx (&a)[5], float sg) {
  const float C1 = 0.30901699437494742f, C2 = -0.80901699437494742f;
  const float S1 = 0.95105651629515357f, S2 = 0.58778525229247313f;
  cpx t1 = cadd(a[1], a[4]), t3 = csub(a[1], a[4]);
  cpx t2 = cadd(a[2], a[3]), t4 = csub(a[2], a[3]);
  cpx y0 = cadd(cadd(a[0], t1), t2);
  cpx m1 = cpx{a[0].r + C1 * t1.r + C2 * t2.r, a[0].i + C1 * t1.i + C2 * t2.i};
  cpx m2 = cpx{a[0].r + C2 * t1.r + C1 * t2.r, a[0].i + C2 * t1.i + C1 * t2.i};
  cpx w1 = cpx{sg * (S1 * t3.r + S2 * t4.r), sg * (S1 * t3.i + S2 * t4.i)};
  cpx w2 = cpx{sg * (S2 * t3.r - S1 * t4.r), sg * (S2 * t3.i - S1 * t4.i)};
  a[0] = y0;
  a[1] = cpx{m1.r - w1.i, m1.i + w1.r};
  a[4] = cpx{m1.r + w1.i, m1.i - w1.r};
  a[2] = cpx{m2.r - w2.i, m2.i + w2.r};
  a[3] = cpx{m2.r + w2.i, m2.i - w2.r};
}

__device__ __forceinline__ void bfly(cpx (&a)[8], float sg) {
  const float C = 0.70710678118654752f;
  cpx t0 = cadd(a[0], a[4]), t1 = csub(a[0], a[4]);
  cpx t2 = cadd(a[2], a[6]), t3 = csub(a[2], a[6]);
  cpx j = cmulj(t3, sg);
  cpx E0 = cadd(t0, t2), E2 = csub(t0, t2);
  cpx E1 = cadd(t1, j),  E3 = csub(t1, j);
  t0 = cadd(a[1], a[5]); t1 = csub(a[1], a[5]);
  t2 = cadd(a[3], a[7]); t3 = csub(a[3], a[7]);
  j = cmulj(t3, sg);
  cpx O0 = cadd(t0, t2), O2 = csub(t0, t2);
  cpx O1 = cadd(t1, j),  O3 = csub(t1, j);
  cpx W1 = cpx{C * (O1.r - sg * O1.i), C * (O1.i + sg * O1.r)};   // *w8
  cpx W2 = cmulj(O2, sg);                                         // *w8^2
  cpx W3 = cpx{-C * (O3.r + sg * O3.i), C * (sg * O3.r - O3.i)};  // *w8^3
  a[0] = cadd(E0, O0); a[4] = csub(E0, O0);
  a[1] = cadd(E1, W1); a[5] = csub(E1, W1);
  a[2] = cadd(E2, W2); a[6] = csub(E2, W2);
  a[3] = cadd(E3, W3); a[7] = csub(E3, W3);
}

// Stockham stage, radix R, ncols interleaved transforms; only stage twiddles
// (R-1 of them) use sincos, the small DFT is constant-folded.
template <int R>
__device__ void stageT(const float* __restrict__ xR, const float* __restrict__ xI,
                       float* __restrict__ yR, float* __restrict__ yI,
                       int N, int l, float sign, int ncols, int tid, int nthr) {
  const int n0 = N / R;
  const int total = n0 * ncols;
  const float wstep = sign * 2.0f * PI_F / (float)(l * R);
  for (int w = tid; w < total; w += nthr) {
    const int c = w % ncols;
    const int idx = w / ncols;
    const int k = idx % l;
    const int bq = idx / l;
    cpx a[R];
    a[0] = cpx{xR[idx * ncols + c], xI[idx * ncols + c]};
    const float wk = wstep * (float)k;
#pragma unroll
    for (int q = 1; q < R; ++q) {
      const int ad = (idx + n0 * q) * ncols + c;
      float s, cc;
      __sincosf(wk * (float)q, &s, &cc);
      const float xr = xR[ad], xi = xI[ad];
      a[q] = cpx{xr * cc - xi * s, xr * s + xi * cc};
    }
    bfly(a, sign);
    const int ob = bq * l * R + k;
#pragma unroll
    for (int p = 0; p < R; ++p) {
      const int oa = (ob + l * p) * ncols + c;
      yR[oa] = a[p].r;
      yI[oa] = a[p].i;
    }
  }
}

// ---------------------------------------------------------------------------
// Radix-16 first Stockham stage (l=1, twiddle-free) of the 1920-pt row FFT on
// the matrix cores: complex DFT-16 as 4 real 16x16x16 matmuls (f32 WMMA, K=4).
// ---------------------------------------------------------------------------
__device__ __forceinline__ void dft16_wmma(const float* __restrict__ sR,
                                           const float* __restrict__ sI,
                                           float* __restrict__ dR,
                                           float* __restrict__ dI,
                                           float sign, int tid) {
  const int lane = tid & 31;
  const int wid  = tid >> 5;
  const int M    = lane & 15;
  const int half = lane >> 4;
  for (int it = 0; it < 2; ++it) {
    const int t = wid + 4 * it;
    const int k0 = t * 16;
    const int kcol = min(k0 + M, 119);
    v8f cRv = {};
    v8f cIv = {};
    for (int kk = 0; kk < 16; kk += 4) {
      const int e0 = kk + 2 * half;
      const int e1 = e0 + 1;
      float s0, c0, s1, c1;
      const int mk0 = (M * e0) & 15;
      const int mk1 = (M * e1) & 15;
      __sincosf(sign * (PI_F / 8.0f) * (float)mk0, &s0, &c0);
      __sincosf(sign * (PI_F / 8.0f) * (float)mk1, &s1, &c1);
      v2f aC;  aC.x = c0;   aC.y = c1;
      v2f aS;  aS.x = s0;   aS.y = s1;
      v2f aSn; aSn.x = -s0; aSn.y = -s1;
      v2f bR; bR.x = sR[e0 * 120 + kcol]; bR.y = sR[e1 * 120 + kcol];
      v2f bI; bI.x = sI[e0 * 120 + kcol]; bI.y = sI[e1 * 120 + kcol];
      cRv = __builtin_amdgcn_wmma_f32_16x16x4_f32(false, aC,  false, bR,
                                                  (short)0, cRv, false, false);
      cRv = __builtin_amdgcn_wmma_f32_16x16x4_f32(false, aSn, false, bI,
                                                  (short)0, cRv, false, false);
      cIv = __builtin_amdgcn_wmma_f32_16x16x4_f32(false, aS,  false, bR,
                                                  (short)0, cIv, false, false);
      cIv = __builtin_amdgcn_wmma_f32_16x16x4_f32(false, aC,  false, bI,
                                                  (short)0, cIv, false, false);
    }
    const int k = k0 + M;
    if (k < 120) {
      for (int v = 0; v < 8; ++v) {
        const int p = v + 8 * half;
        dR[k * 16 + p] = cRv[v];
        dI[k * 16 + p] = cIv[v];
      }
    }
  }
}

#define SWAPP(pa, pb) { float* _t = (pa); (pa) = (pb); (pb) = _t; }

// ---------------------------------------------------------------------------
// Forward row FFT (1920 = 16*8*3*5) with pad+ifftshift+V_div*mu1 fused load.
// ---------------------------------------------------------------------------
__global__ __launch_bounds__(128) void k_fft_rows_fwd(
    const float* __restrict__ src, const float* __restrict__ scalePtr,
    float* __restrict__ outR, float* __restrict__ outI, int batched) {
  extern __shared__ float smem[];
  float* AR = smem;
  float* AI = AR + FW;
  float* BR = AI + FW;
  float* BI = BR + FW;
  const int yy = blockIdx.x;
  const int b  = blockIdx.y;
  const int ys = (810 + yy) % FH;

  float scale = 1.0f;
  if (scalePtr) scale = scalePtr[4] / (1.0f + MU1_0);
  const float* srcRow = src + ((size_t)(batched ? b : 0) * SH + yy) * SW;

  for (int xs = threadIdx.x; xs < FW; xs += blockDim.x) {
    int x0 = xs + SW; if (x0 >= FW) x0 -= FW;
    const int sc = x0 - 480;
    float v = 0.0f;
    if (sc >= 0 && sc < SW) v = srcRow[sc] * scale;
    AR[xs] = v;
    AI[xs] = 0.0f;
  }
  __syncthreads();
  dft16_wmma(AR, AI, BR, BI, -1.0f, threadIdx.x);
  __syncthreads();
  stageT<8>(BR, BI, AR, AI, FW, 16,  -1.0f, 1, threadIdx.x, blockDim.x);
  __syncthreads();
  stageT<3>(AR, AI, BR, BI, FW, 128, -1.0f, 1, threadIdx.x, blockDim.x);
  __syncthreads();
  stageT<5>(BR, BI, AR, AI, FW, 384, -1.0f, 1, threadIdx.x, blockDim.x);
  __syncthreads();

  const size_t base = ((size_t)b * FH + ys) * FW;
#ifdef HAVE_TDM
  if ((threadIdx.x >> 5) == 0) {
    tdm_store_tile(&outR[base], lds_off(smem, AR), FW, 1, FW);
    tdm_store_tile(&outI[base], lds_off(smem, AI), FW, 1, FW);
    __builtin_amdgcn_s_wait_tensorcnt(0);
  }
#else
  for (int i = threadIdx.x; i < FW; i += blockDim.x) {
    outR[base + i] = AR[i];
    outI[base + i] = AI[i];
  }
#endif
}

// ---------------------------------------------------------------------------
// Inverse row FFT with fftshift+crop+real() fused store.
// ---------------------------------------------------------------------------
__global__ __launch_bounds__(128) void k_fft_rows_inv_crop(
    const float* __restrict__ ZR, const float* __restrict__ ZI,
    float* __restrict__ out) {
  extern __shared__ float smem[];
  float* AR = smem;
  float* AI = AR + FW;
  float* BR = AI + FW;
  float* BI = BR + FW;
  const int yy = blockIdx.x;
  const int b  = blockIdx.y;
  const int ys = (810 + yy) % FH;
  const size_t base = ((size_t)b * FH + ys) * FW;

#ifdef HAVE_TDM
  if ((threadIdx.x >> 5) == 0) {
    tdm_load_tile(&ZR[base], lds_off(smem, AR), FW, 1, FW);
    tdm_load_tile(&ZI[base], lds_off(smem, AI), FW, 1, FW);
    __builtin_amdgcn_s_wait_tensorcnt(0);
  }
#else
  for (int i = threadIdx.x; i < FW; i += blockDim.x) {
    AR[i] = ZR[base + i];
    AI[i] = ZI[base + i];
  }
#endif
  __syncthreads();
  dft16_wmma(AR, AI, BR, BI, +1.0f, threadIdx.x);
  __syncthreads();
  stageT<8>(BR, BI, AR, AI, FW, 16,  +1.0f, 1, threadIdx.x, blockDim.x);
  __syncthreads();
  stageT<3>(AR, AI, BR, BI, FW, 128, +1.0f, 1, threadIdx.x, blockDim.x);
  __syncthreads();
  stageT<5>(BR, BI, AR, AI, FW, 384, +1.0f, 1, threadIdx.x, blockDim.x);
  __syncthreads();

  for (int xx = threadIdx.x; xx < SW; xx += blockDim.x) {
    const int xs = (1440 + xx) % FW;
    out[((size_t)b * SH + yy) * SW + xx] = AR[xs];
  }
}

// ---------------------------------------------------------------------------
// Column pass (1080 = 8*3*3*3*5), TC columns per block in LDS.
// roundtrip==0: forward only (Hf).  roundtrip==1: fwd FFT -> *K -> inv FFT,
// one LDS residency (saves a full 530MB global round-trip vs separate passes).
// ---------------------------------------------------------------------------
__global__ __launch_bounds__(256) void k_fft_cols(
    float* __restrict__ R, float* __restrict__ I, float sign,
    const float* __restrict__ KR, const float* __restrict__ KI, int roundtrip) {
  extern __shared__ float smem[];
  float* AR = smem;
  float* AI = AR + FH * TC;
  float* BR = AI + FH * TC;
  float* BI = BR + FH * TC;
  const int c0 = blockIdx.x * TC;
  const int b  = blockIdx.y;
  const size_t base = (size_t)b * FH * FW + c0;
  const int tid = threadIdx.x, nt = blockDim.x;

#ifdef HAVE_TDM
  if ((tid >> 5) == 0) {
    tdm_load_tile(&R[base], lds_off(smem, AR), TC, FH, FW);
    tdm_load_tile(&I[base], lds_off(smem, AI), TC, FH, FW);
    __builtin_amdgcn_s_wait_tensorcnt(0);
  }
#else
  for (int i = tid; i < FH * TC; i += nt) {
    const size_t g = base + (size_t)(i / TC) * FW + (i % TC);
    __builtin_prefetch(&R[g + (size_t)64 * FW], 0, 1);
    AR[i] = R[g];
    AI[i] = I[g];
  }
#endif
  __syncthreads();

  float *cr = AR, *ci = AI, *dr = BR, *di = BI;
  stageT<8>(cr, ci, dr, di, FH, 1,   sign, TC, tid, nt); __syncthreads(); SWAPP(cr, dr); SWAPP(ci, di);
  stageT<3>(cr, ci, dr, di, FH, 8,   sign, TC, tid, nt); __syncthreads(); SWAPP(cr, dr); SWAPP(ci, di);
  stageT<3>(cr, ci, dr, di, FH, 24,  sign, TC, tid, nt); __syncthreads(); SWAPP(cr, dr); SWAPP(ci, di);
  stageT<3>(cr, ci, dr, di, FH, 72,  sign, TC, tid, nt); __syncthreads(); SWAPP(cr, dr); SWAPP(ci, di);
  stageT<5>(cr, ci, dr, di, FH, 216, sign, TC, tid, nt); __syncthreads(); SWAPP(cr, dr); SWAPP(ci, di);

  if (roundtrip) {
    // pointwise K (conj(Hf)*R_div*norm already folded in by k_prepK)
    for (int i = tid; i < FH * TC; i += nt) {
      const size_t p = (size_t)(i / TC) * FW + c0 + (i % TC);
      const float kr = KR[p], ki = KI[p];
      const float zr = cr[i], zi = ci[i];
      cr[i] = zr * kr - zi * ki;
      ci[i] = zr * ki + zi * kr;
    }
    __syncthreads();
    const float isg = -sign;
    stageT<8>(cr, ci, dr, di, FH, 1,   isg, TC, tid, nt); __syncthreads(); SWAPP(cr, dr); SWAPP(ci, di);
    stageT<3>(cr, ci, dr, di, FH, 8,   isg, TC, tid, nt); __syncthreads(); SWAPP(cr, dr); SWAPP(ci, di);
    stageT<3>(cr, ci, dr, di, FH, 24,  isg, TC, tid, nt); __syncthreads(); SWAPP(cr, dr); SWAPP(ci, di);
    stageT<3>(cr, ci, dr, di, FH, 72,  isg, TC, tid, nt); __syncthreads(); SWAPP(cr, dr); SWAPP(ci, di);
    stageT<5>(cr, ci, dr, di, FH, 216, isg, TC, tid, nt); __syncthreads(); SWAPP(cr, dr); SWAPP(ci, di);
  }

#ifdef HAVE_TDM
  if ((tid >> 5) == 0) {
    tdm_store_tile(&R[base], lds_off(smem, cr), TC, FH, FW);
    tdm_store_tile(&I[base], lds_off(smem, ci), TC, FH, FW);
    __builtin_amdgcn_s_wait_tensorcnt(0);
  }
#else
  for (int i = tid; i < FH * TC; i += nt) {
    const size_t g = base + (size_t)(i / TC) * FW + (i % TC);
    R[g] = cr[i];
    I[g] = ci[i];
  }
#endif
}

// ---------------------------------------------------------------------------
// Hf -> K = conj(Hf) * R_div / (FH*FW), in place.
// ---------------------------------------------------------------------------
__global__ __launch_bounds__(256) void k_prepK(float* __restrict__ HR,
                                               float* __restrict__ HI) {
  const size_t p = (size_t)blockIdx.x * blockDim.x + threadIdx.x;
  if (p >= (size_t)FH * FW) return;
  const int y = (int)(p / FW);
  const int x = (int)(p % FW);
  const float hr = HR[p], hi = HI[p];
  const float psps = 4.0f - 2.0f * __cosf(2.0f * PI_F * (float)y / (float)FH)
                          - 2.0f * __cosf(2.0f * PI_F * (float)x / (float)FW);
  const float rdiv = 1.0f / (MU1_0 * (hr * hr + hi * hi) + MU2_0 * psps + MU3_0);
  const float s = rdiv * (1.0f / ((float)FH * (float)FW));
  HR[p] = hr * s;
  HI[p] = -hi * s;
}

// ---------------------------------------------------------------------------
extern "C" void kernel_launch(void* const* d_in, const int* in_sizes, int n_in,
                              void* d_out, int out_size, void* d_ws, size_t ws_size,
                              hipStream_t stream) {
  (void)in_sizes; (void)n_in; (void)out_size; (void)ws_size;
  const float* x   = (const float*)d_in[0];
  const float* h   = (const float*)d_in[1];
  const float* mu1 = (const float*)d_in[2];
  float* out = (float*)d_out;

  const size_t plane = (size_t)FH * FW;
  float* ZR = (float*)d_ws;
  float* ZI = ZR + (size_t)NB * plane;
  float* HR = ZI + (size_t)NB * plane;
  float* HI = HR + plane;

  hipMemsetAsync(d_ws, 0, ((size_t)2 * NB + 2) * plane * sizeof(float), stream);

  const size_t rowLds = (size_t)4 * FW * sizeof(float);       // 30.7 KB
  const size_t colLds = (size_t)4 * FH * TC * sizeof(float);  // 138 KB

  // K = conj(fft2(ifftshift(pad(h)))) * R_div / (FH*FW)
  k_fft_rows_fwd<<<dim3(SH, 1), 128, rowLds, stream>>>(h, nullptr, HR, HI, 0);
  k_fft_cols<<<dim3(FW / TC, 1), 256, colLds, stream>>>(HR, HI, -1.0f,
                                                        nullptr, nullptr, 0);
  k_prepK<<<(unsigned)((plane + 255) / 256), 256, 0, stream>>>(HR, HI);

  // Z = row-FFT(pad(x)*mu1[4]*V_div); then fused col-FFT -> *K -> col-IFFT.
  k_fft_rows_fwd<<<dim3(SH, NB), 128, rowLds, stream>>>(x, mu1, ZR, ZI, 1);
  k_fft_cols<<<dim3(FW / TC, NB), 256, colLds, stream>>>(ZR, ZI, -1.0f, HR, HI, 1);

  // out = crop(fftshift(real(row-IFFT(Z))))
  k_fft_rows_inv_crop<<<dim3(SH, NB), 128, rowLds, stream>>>(ZR, ZI, out);
}